// HungarianMatcher_48026324304468
// MI455X (gfx1250) — compile-verified
//
#include <hip/hip_runtime.h>
#include <math.h>

#define NSLOTS 8192
#define NTGT   2048
#define NCLS   91
#define BM     128   // slots per block (grid.y)
#define BN     256   // targets per block (grid.x)

typedef __attribute__((ext_vector_type(2)))  float        v2f;
typedef __attribute__((ext_vector_type(8)))  float        v8f;
typedef __attribute__((ext_vector_type(4)))  unsigned int u32x4;
typedef __attribute__((ext_vector_type(8)))  int          i32x8;
typedef __attribute__((ext_vector_type(4)))  int          i32x4;

// ---------------------------------------------------------------------------
// Kernel 1: per-row log-sum-exp of logits [NSLOTS x NCLS].
// One wave handles 16 rows. Row max via VALU + xor-16 shuffle; sum of exp via
// V_WMMA_F32_16X16X4_F32 against an all-ones B matrix (D[m,*] = row sums),
// 23 accumulating WMMA issues covering K = 0..91 (tail lane zero-padded).
// A-matrix layout (ISA 7.12.2, 32-bit A 16x4): lanes 0-15 hold K={0,1},
// lanes 16-31 hold K={2,3}, row M = lane % 16.
// ---------------------------------------------------------------------------
__global__ __launch_bounds__(256) void lse_kernel(const float* __restrict__ logits,
                                                  float* __restrict__ lse) {
  const int lane  = threadIdx.x & 31;
  const int wave  = threadIdx.x >> 5;
  const int row0  = (blockIdx.x * 8 + wave) * 16;   // 16 rows per wave
  const int m     = lane & 15;                      // row within 16-row tile
  const int khalf = lane >> 4;                      // 0 -> K{0,1}, 1 -> K{2,3}
  const float* rowp = logits + (size_t)(row0 + m) * NCLS;

  // ---- pass 1: row max (each lane covers half of each 4-wide K chunk) ----
  float mx = -INFINITY;
  #pragma unroll
  for (int kk = 0; kk < 23; ++kk) {
    int c0 = 4 * kk + 2 * khalf;
    if (c0     < NCLS) mx = fmaxf(mx, rowp[c0]);
    if (c0 + 1 < NCLS) mx = fmaxf(mx, rowp[c0 + 1]);
  }
  mx = fmaxf(mx, __shfl_xor(mx, 16, 32));           // combine K-halves per row

  // ---- pass 2: sum(exp(x - max)) via WMMA against ones ----
  v8f acc = {};
  v2f ones; ones.x = 1.0f; ones.y = 1.0f;
  #pragma unroll
  for (int kk = 0; kk < 23; ++kk) {
    int c0 = 4 * kk + 2 * khalf;
    v2f a;
    a.x = (c0     < NCLS) ? __expf(rowp[c0]     - mx) : 0.0f;
    a.y = (c0 + 1 < NCLS) ? __expf(rowp[c0 + 1] - mx) : 0.0f;
    acc = __builtin_amdgcn_wmma_f32_16x16x4_f32(false, a, false, ones,
                                                (short)0, acc, false, false);
  }

  // D layout: VGPR r -> row r (lanes 0-15) / row r+8 (lanes 16-31), N = lane%16.
  // lanes 0..7 own rows 0..7 (acc[lane]); lanes 16..23 own rows 8..15.
  int   orow = (lane < 16) ? lane : (lane - 8);               // output row idx
  float mxo  = __shfl(mx, (lane >= 16) ? (lane - 8) : lane, 32);
  if (lane < 8 || (lane >= 16 && lane < 24)) {
    float s = acc[orow & 7];
    lse[row0 + orow] = mxo + __logf(s);
  }
}

// ---------------------------------------------------------------------------
// TDM: 1-D contiguous DMA of `ndw` dwords from global to LDS.
// Descriptor per ISA ch.8: group0 = {count=1, lds_addr, global_addr, type=2},
// group1 = {data_size=4B, tensor_dim0=ndw, tensor_dim1=1, tile_dim0=ndw,
// tile_dim1=tile_dim2=1, strides=ndw}. Groups 2/3 zero (<=2D tensor).
// Issued per-wave (EXEC ignored); tracked with TENSORcnt.
// ---------------------------------------------------------------------------
__device__ __forceinline__ void tdm_load_dwords(unsigned lds_addr,
                                                const void* gsrc,
                                                unsigned ndw) {
  unsigned long long ga = (unsigned long long)(uintptr_t)gsrc;
  u32x4 g0;
  g0.x = 1u;                                          // count = 1 (valid)
  g0.y = lds_addr;                                    // LDS byte address
  g0.z = (unsigned)(ga & 0xFFFFFFFFu);                // global_addr[31:0]
  g0.w = (unsigned)((ga >> 32) & 0x01FFFFFFu)         // global_addr[56:32]
         | 0x80000000u;                               // type = 2 ("image")
  i32x8 g1;
  g1[0] = (int)(2u << 16);                            // data_size = 4 bytes
  g1[1] = (int)((ndw & 0xFFFFu) << 16);               // tensor_dim0[15:0]
  g1[2] = (int)((ndw >> 16) | (1u << 16));            // dim0[31:16] | tensor_dim1=1
  g1[3] = (int)((ndw & 0xFFFFu) << 16);               // dim1[31:16]=0 | tile_dim0
  g1[4] = (int)(1u | (1u << 16));                     // tile_dim1=1 | tile_dim2=1
  g1[5] = (int)ndw;                                   // tensor_dim0_stride[31:0]
  g1[6] = (int)((ndw & 0xFFFFu) << 16);               // stride0[47:32]=0 | stride1[15:0]
  g1[7] = (int)(ndw >> 16);                           // stride1[47:16]
  i32x4 gz = {};
#if defined(__clang_major__) && __clang_major__ >= 23
  i32x8 z8 = {};
  __builtin_amdgcn_tensor_load_to_lds(g0, g1, gz, gz, z8, 0);
#else
  __builtin_amdgcn_tensor_load_to_lds(g0, g1, gz, gz, 0);
#endif
}

// ---------------------------------------------------------------------------
// Kernel 2: fused cost-matrix tile. Block = 256 threads (8 wave32),
// tile = BM(128) slots x BN(256) targets; one target column per thread.
// Wave 0 DMAs the target tile (boxes + ids) into LDS via TDM while waves 1-4
// stage per-slot row data (cxcywh, xyxy, area, lse) into LDS.
// Inner loop: slot data is block-uniform (LDS broadcast), each thread emits
// one NT store per slot; writes per iteration are a contiguous 1 KB span.
// ---------------------------------------------------------------------------
__global__ __launch_bounds__(256) void cost_kernel(const float* __restrict__ logits,
                                                   const float* __restrict__ pred_boxes,
                                                   const float* __restrict__ tgt_boxes,
                                                   const int*   __restrict__ tgt_ids,
                                                   const float* __restrict__ lse,
                                                   float*       __restrict__ out) {
  __shared__ __align__(16) float s_tbox[BN * 4];     // target cxcywh (TDM)
  __shared__ __align__(16) int   s_tid[BN];          // target class ids (TDM)
  __shared__ __align__(16) float s_slot[BM * 12];    // cx,cy,w,h,x0,y0,x1,y1,area,lse,--,--

  const int t  = threadIdx.x;
  const int m0 = blockIdx.x * BN;
  const int n0 = blockIdx.y * BM;

  if ((t >> 5) == 0) {
    // wave 0: async-tensor DMA of the target tile, then wait TENSORcnt==0
    tdm_load_dwords((unsigned)(uintptr_t)&s_tbox[0], tgt_boxes + (size_t)m0 * 4, BN * 4);
    tdm_load_dwords((unsigned)(uintptr_t)&s_tid[0],  tgt_ids + m0,               BN);
    __builtin_amdgcn_s_wait_tensorcnt(0);
  } else if (t >= 32 && t < 32 + BM) {
    // waves 1-4: stage per-slot row data (overlaps with the TDM)
    int s = t - 32;
    float4 b = *(const float4*)(pred_boxes + (size_t)(n0 + s) * 4);
    float x0 = b.x - 0.5f * b.z, y0 = b.y - 0.5f * b.w;
    float x1 = b.x + 0.5f * b.z, y1 = b.y + 0.5f * b.w;
    float* sp = &s_slot[s * 12];
    sp[0] = b.x; sp[1] = b.y; sp[2] = b.z; sp[3] = b.w;
    sp[4] = x0;  sp[5] = y0;  sp[6] = x1;  sp[7] = y1;
    sp[8] = (x1 - x0) * (y1 - y0);
    sp[9] = lse[n0 + s];
    sp[10] = 0.0f; sp[11] = 0.0f;
  }
  __syncthreads();

  // per-thread target registers
  float4 tb = *(const float4*)&s_tbox[t * 4];
  const float tcx = tb.x, tcy = tb.y, tw = tb.z, th_ = tb.w;
  const float tx0 = tcx - 0.5f * tw, ty0 = tcy - 0.5f * th_;
  const float tx1 = tcx + 0.5f * tw, ty1 = tcy + 0.5f * th_;
  const float tarea = (tx1 - tx0) * (ty1 - ty0);
  const int   tcls  = s_tid[t];

  const float* lg = logits + (size_t)n0 * NCLS + tcls;   // gather column, step NCLS
  float*       op = out + (size_t)n0 * NTGT + (m0 + t);  // output column, step NTGT

  #pragma unroll 4
  for (int i = 0; i < BM; ++i) {
    const float* sp = &s_slot[i * 12];
    float4 pa = *(const float4*)(sp);       // cx cy w h
    float4 pb = *(const float4*)(sp + 4);   // x0 y0 x1 y1
    float  parea = sp[8], plse = sp[9];

    // L1 on raw cxcywh
    float l1 = fabsf(pa.x - tcx) + fabsf(pa.y - tcy) +
               fabsf(pa.z - tw)  + fabsf(pa.w - th_);

    // GIoU
    float ix0 = fmaxf(pb.x, tx0), iy0 = fmaxf(pb.y, ty0);
    float ix1 = fminf(pb.z, tx1), iy1 = fminf(pb.w, ty1);
    float inter = fmaxf(ix1 - ix0, 0.0f) * fmaxf(iy1 - iy0, 0.0f);
    float uni   = parea + tarea - inter;
    float iou   = inter / uni;
    float cx0 = fminf(pb.x, tx0), cy0 = fminf(pb.y, ty0);
    float cx1 = fmaxf(pb.z, tx1), cy1 = fmaxf(pb.w, ty1);
    float ac  = fmaxf(cx1 - cx0, 0.0f) * fmaxf(cy1 - cy0, 0.0f);
    float giou = iou - (ac - uni) / ac;

    // class cost: -softmax prob via precomputed log-sum-exp
    float prob = __expf(lg[(size_t)i * NCLS] - plse);

    float cost = l1 - prob - giou;
    __builtin_nontemporal_store(cost, op + (size_t)i * NTGT);
  }
}

// ---------------------------------------------------------------------------
extern "C" void kernel_launch(void* const* d_in, const int* in_sizes, int n_in,
                              void* d_out, int out_size, void* d_ws, size_t ws_size,
                              hipStream_t stream) {
  const float* logits     = (const float*)d_in[0];  // [8192, 91]
  const float* pred_boxes = (const float*)d_in[1];  // [8192, 4]
  const float* tgt_boxes  = (const float*)d_in[2];  // [2048, 4]
  const int*   tgt_ids    = (const int*)d_in[3];    // [2048]
  float*       out        = (float*)d_out;          // [8192, 2048]
  float*       lse        = (float*)d_ws;           // [8192] scratch

  lse_kernel<<<NSLOTS / 128, 256, 0, stream>>>(logits, lse);

  dim3 grid(NTGT / BN, NSLOTS / BM);
  cost_kernel<<<grid, 256, 0, stream>>>(logits, pred_boxes, tgt_boxes, tgt_ids, lse, out);

  (void)in_sizes; (void)n_in; (void)out_size; (void)ws_size;
}